// _SA_3908420240282
// MI455X (gfx1250) — compile-verified
//
#include <hip/hip_runtime.h>
#include <hip/hip_bf16.h>

// Set-abstraction layer: FPS -> ball-query(top-32 nearest within radius) ->
// gather -> MLP(67->64->128, exact GELU) with fp32 WMMA -> max-pool over K.

#define N_PTS   32768
#define M_CENT  2048
#define KSAMP   32
#define D_IN    64
#define D0P     68     /* 67 padded to multiple of 4 for 16x16x4 f32 WMMA */
#define D1      64
#define D2      128
#define R2      0.25f
#define CAND_CAP 4096

typedef float v2f __attribute__((ext_vector_type(2)));
typedef float v8f __attribute__((ext_vector_type(8)));

// ---------------------------------------------------------------------------
// Kernel 1: farthest point sampling. Single persistent workgroup; dist[] in
// LDS (128KB of the 320KB WGP pool). Argmax tie-break = lowest index, matching
// jnp.argmax, via packed key (distBits<<32 | ~idx) and a max-reduction.
// ---------------------------------------------------------------------------
__global__ __launch_bounds__(1024) void _sa_fps_kernel(
    const float* __restrict__ xyz, int* __restrict__ idx_cent,
    float* __restrict__ new_xyz) {
  __shared__ float dist[N_PTS];
  __shared__ unsigned long long red[32];
  __shared__ int s_far;

  const int tid = threadIdx.x;
  for (int j = tid; j < N_PTS; j += 1024) dist[j] = 1e38f;
  if (tid == 0) s_far = 0;
  __syncthreads();

  for (int i = 0; i < M_CENT; ++i) {
    const int far = s_far;
    const float fx = xyz[3 * far + 0];
    const float fy = xyz[3 * far + 1];
    const float fz = xyz[3 * far + 2];
    if (tid == 0) {
      idx_cent[i] = far;
      new_xyz[3 * i + 0] = fx;
      new_xyz[3 * i + 1] = fy;
      new_xyz[3 * i + 2] = fz;
    }
    unsigned long long best = 0ull;
    for (int j = tid; j < N_PTS; j += 1024) {
      const float dx = xyz[3 * j + 0] - fx;
      const float dy = xyz[3 * j + 1] - fy;
      const float dz = xyz[3 * j + 2] - fz;
      const float d = dx * dx + dy * dy + dz * dz;
      const float nd = fminf(dist[j], d);
      dist[j] = nd;
      const unsigned long long key =
          ((unsigned long long)__float_as_uint(nd) << 32) |
          (unsigned)(~(unsigned)j);
      if (key > best) best = key;
    }
    // wave32 max-reduction
    for (int off = 16; off > 0; off >>= 1) {
      const unsigned long long o = __shfl_down(best, off, 32);
      if (o > best) best = o;
    }
    if ((tid & 31) == 0) red[tid >> 5] = best;
    __syncthreads();
    if (tid == 0) {
      unsigned long long b = red[0];
      for (int w = 1; w < 32; ++w)
        if (red[w] > b) b = red[w];
      s_far = (int)(~(unsigned)(b & 0xFFFFFFFFull));
    }
    __syncthreads();
  }
}

// ---------------------------------------------------------------------------
// Kernel 2: ball query == top-32 smallest of dist2 masked to 1e9 outside r.
// Only the SET of 32 indices matters (max-pool downstream). Compact in-radius
// candidates to LDS, then 32 exact min-extractions on (d2bits,idx) keys.
// If <32 in radius, pad with lowest-index out-of-radius points (the 1e9 ties).
// ---------------------------------------------------------------------------
__global__ __launch_bounds__(256) void _sa_ballq_kernel(
    const float* __restrict__ xyz, const float* __restrict__ new_xyz,
    int* __restrict__ group_idx) {
  __shared__ unsigned long long cand[CAND_CAP];
  __shared__ unsigned long long red[8];
  __shared__ unsigned long long s_best;
  __shared__ int s_cnt;

  const int m = blockIdx.x;
  const int tid = threadIdx.x;
  const float cx = new_xyz[3 * m + 0];
  const float cy = new_xyz[3 * m + 1];
  const float cz = new_xyz[3 * m + 2];
  if (tid == 0) s_cnt = 0;
  __syncthreads();

  for (int j = tid; j < N_PTS; j += 256) {
    const float dx = xyz[3 * j + 0] - cx;
    const float dy = xyz[3 * j + 1] - cy;
    const float dz = xyz[3 * j + 2] - cz;
    const float d2 = dx * dx + dy * dy + dz * dz;
    if (d2 <= R2) {
      const int p = atomicAdd(&s_cnt, 1);
      if (p < CAND_CAP)
        cand[p] = ((unsigned long long)__float_as_uint(d2) << 32) | (unsigned)j;
    }
  }
  __syncthreads();
  const int cnt = min(s_cnt, CAND_CAP);
  int* out = group_idx + m * KSAMP;

  if (cnt >= KSAMP) {
    for (int s = 0; s < KSAMP; ++s) {
      unsigned long long best = ~0ull;
      for (int p = tid; p < cnt; p += 256) {
        const unsigned long long k = cand[p];
        if (k < best) best = k;
      }
      for (int off = 16; off > 0; off >>= 1) {
        const unsigned long long o = __shfl_down(best, off, 32);
        if (o < best) best = o;
      }
      if ((tid & 31) == 0) red[tid >> 5] = best;
      __syncthreads();
      if (tid == 0) {
        unsigned long long b = red[0];
        for (int w = 1; w < 8; ++w)
          if (red[w] < b) b = red[w];
        s_best = b;
        out[s] = (int)(b & 0xFFFFFFFFull);
      }
      __syncthreads();
      const unsigned long long b = s_best;
      for (int p = tid; p < cnt; p += 256)
        if (cand[p] == b) cand[p] = ~0ull;
      __syncthreads();
    }
  } else {
    for (int p = tid; p < cnt; p += 256)
      out[p] = (int)(cand[p] & 0xFFFFFFFFull);
    __syncthreads();
    if (tid < 32) {  // wave 0 pads with lowest-index out-of-radius points
      int fill = cnt;
      int base = 0;
      while (fill < KSAMP && base < N_PTS) {
        const int j = base + tid;
        bool farpt = false;
        if (j < N_PTS) {
          const float dx = xyz[3 * j + 0] - cx;
          const float dy = xyz[3 * j + 1] - cy;
          const float dz = xyz[3 * j + 2] - cz;
          farpt = (dx * dx + dy * dy + dz * dz) > R2;
        }
        const unsigned mask = (unsigned)__ballot(farpt);
        const int prefix = __popc(mask & ((1u << tid) - 1u));
        if (farpt && (fill + prefix) < KSAMP) out[fill + prefix] = j;
        fill += __popc(mask);
        base += 32;
      }
    }
  }
}

// ---------------------------------------------------------------------------
// Kernel 3: gather + MLP with fp32 WMMA (V_WMMA_F32_16X16X4_F32) + max-pool.
// One workgroup (8 waves) per centroid: rows = 32 samples.
// Phase 1: A[32x68] (xyz-delta ++ feat, K padded) x W1[68x64] -> GELU -> sH.
// Phase 2: sH[32x64] x W2[64x128] -> GELU -> max over 32 rows -> pooled.
// LDS buffer `buf` is reused across phases (A+W1 then W2+Out).
// ---------------------------------------------------------------------------
__global__ __launch_bounds__(256) void _sa_mlp_kernel(
    const float* __restrict__ xyz, const float* __restrict__ feat,
    const float* __restrict__ W1, const float* __restrict__ b1,
    const float* __restrict__ W2, const float* __restrict__ b2,
    const float* __restrict__ new_xyz, const int* __restrict__ group_idx,
    float* __restrict__ pooled) {
  __shared__ float buf[D1 * D2 + KSAMP * D2];  // phase2 is the larger phase
  __shared__ float sH[KSAMP * D1];
  __shared__ float sB1[D1];
  __shared__ float sB2[D2];
  __shared__ int sIdx[KSAMP];
  __shared__ float sC[3];

  const int m = blockIdx.x;
  const int tid = threadIdx.x;
  const int wave = tid >> 5;
  const int lane = tid & 31;

  if (tid < KSAMP) sIdx[tid] = group_idx[m * KSAMP + tid];
  if (tid < 3) sC[tid] = new_xyz[3 * m + tid];
  if (tid >= 32 && tid < 96) sB1[tid - 32] = b1[tid - 32];
  if (tid >= 96 && tid < 224) sB2[tid - 96] = b2[tid - 96];
  __syncthreads();

  float* sA = buf;                   // [32][68]
  float* sW1 = buf + KSAMP * D0P;    // [68][64]
  for (int e = tid; e < KSAMP * D0P; e += 256) {
    const int r = e / D0P;
    const int c = e - r * D0P;
    const int idx = sIdx[r];
    float v;
    if (c < 3)        v = xyz[3 * idx + c] - sC[c];
    else if (c < 67)  v = feat[idx * D_IN + (c - 3)];
    else              v = 0.0f;  // K padding
    sA[e] = v;
  }
  for (int e = tid; e < D0P * D1; e += 256)
    sW1[e] = (e < 67 * D1) ? W1[e] : 0.0f;  // zero-pad row 67
  __syncthreads();

  // ---- layer 1: 2 row-tiles x 4 col-tiles = 8 tiles, one per wave ----
  {
    const int tr = wave >> 2;           // 0..1
    const int tc = wave & 3;            // 0..3
    const int mrow = tr * 16 + (lane & 15);
    const int ncol = tc * 16 + (lane & 15);
    const int khalf = (lane >> 4) * 2;  // lanes 0-15 -> K{0,1}, 16-31 -> K{2,3}
    v8f acc = {};
    for (int kk = 0; kk < D0P / 4; ++kk) {
      const int kb = kk * 4 + khalf;
      v2f a, b;
      a.x = sA[mrow * D0P + kb];
      a.y = sA[mrow * D0P + kb + 1];
      b.x = sW1[kb * D1 + ncol];
      b.y = sW1[(kb + 1) * D1 + ncol];
      acc = __builtin_amdgcn_wmma_f32_16x16x4_f32(false, a, false, b,
                                                  (short)0, acc, false, false);
    }
    for (int j = 0; j < 8; ++j) {
      const int row = tr * 16 + j + (lane >> 4) * 8;
      const int col = tc * 16 + (lane & 15);
      const float x = acc[j] + sB1[col];
      sH[row * D1 + col] = 0.5f * x * (1.0f + erff(x * 0.70710678118654752f));
    }
  }
  __syncthreads();

  // ---- phase 2 staging: reuse buf for W2 + Out ----
  float* sW2 = buf;                 // [64][128]
  float* sOut = buf + D1 * D2;      // [32][128]
  for (int e = tid; e < D1 * D2; e += 256) sW2[e] = W2[e];
  __syncthreads();

  // ---- layer 2: 2 row-tiles x 8 col-tiles = 16 tiles, two per wave ----
  for (int t = wave; t < 16; t += 8) {
    const int tr = t >> 3;           // 0..1
    const int tc = t & 7;            // 0..7
    const int mrow = tr * 16 + (lane & 15);
    const int ncol = tc * 16 + (lane & 15);
    const int khalf = (lane >> 4) * 2;
    v8f acc = {};
    for (int kk = 0; kk < D1 / 4; ++kk) {
      const int kb = kk * 4 + khalf;
      v2f a, b;
      a.x = sH[mrow * D1 + kb];
      a.y = sH[mrow * D1 + kb + 1];
      b.x = sW2[kb * D2 + ncol];
      b.y = sW2[(kb + 1) * D2 + ncol];
      acc = __builtin_amdgcn_wmma_f32_16x16x4_f32(false, a, false, b,
                                                  (short)0, acc, false, false);
    }
    for (int j = 0; j < 8; ++j) {
      const int row = tr * 16 + j + (lane >> 4) * 8;
      const int col = tc * 16 + (lane & 15);
      const float x = acc[j] + sB2[col];
      sOut[row * D2 + col] = 0.5f * x * (1.0f + erff(x * 0.70710678118654752f));
    }
  }
  __syncthreads();

  // ---- max-pool over K=32 samples ----
  if (tid < D2) {
    float mx = -3.4e38f;
    for (int s = 0; s < KSAMP; ++s) mx = fmaxf(mx, sOut[s * D2 + tid]);
    pooled[m * D2 + tid] = mx;
  }
}

// ---------------------------------------------------------------------------
extern "C" void kernel_launch(void* const* d_in, const int* in_sizes, int n_in,
                              void* d_out, int out_size, void* d_ws,
                              size_t ws_size, hipStream_t stream) {
  (void)in_sizes; (void)n_in; (void)out_size; (void)ws_size;
  const float* xyz  = (const float*)d_in[0];
  const float* feat = (const float*)d_in[1];
  const float* W1   = (const float*)d_in[2];
  const float* b1   = (const float*)d_in[3];
  const float* W2   = (const float*)d_in[4];
  const float* b2   = (const float*)d_in[5];

  float* out = (float*)d_out;
  float* new_xyz = out;                 // [2048,3]
  float* pooled = out + M_CENT * 3;     // [2048,128]

  int* idx_cent  = (int*)d_ws;          // [2048]
  int* group_idx = idx_cent + M_CENT;   // [2048,32]

  _sa_fps_kernel<<<1, 1024, 0, stream>>>(xyz, idx_cent, new_xyz);
  _sa_ballq_kernel<<<M_CENT, 256, 0, stream>>>(xyz, new_xyz, group_idx);
  _sa_mlp_kernel<<<M_CENT, 256, 0, stream>>>(xyz, feat, W1, b1, W2, b2,
                                             new_xyz, group_idx, pooled);
}